// RNN_87557203296509
// MI455X (gfx1250) — compile-verified
//
#include <hip/hip_runtime.h>
#include <hip/hip_bf16.h>
#include <math.h>

// ---------------------------------------------------------------------------
// RNN(2350/768) + MHA + scan collapse, fp32 WMMA (v_wmma_f32_16x16x4_f32).
//
// Key restructurings (exact algebra, not approximations):
//  * scan2's carry never feeds the cells -> final state = cell output at
//    t_b = seq_len[b]-1 -> scan2 becomes one 64-row cell evaluation.
//  * MHA needs only query row t_b per batch; K/V projections folded:
//      scores = (Wk^T q) . h1s[s]   (softmax-invariant constant dropped)
//      ctx    = Wv (sum_s a_s h1s[s]) + bv   (softmax sums to 1)
//  * scan1 per-step GEMMs use concatenated-K (x-term + h-term) so no X1
//    precompute buffer is needed; Whh1/Wih1 (~29MB) live in the 192MB L2.
//  * GEMM inner loop: unguarded float2 (b64) loads over K&~3, single guarded
//    tail iteration -> no EXEC save/restore in the hot loop.
// ---------------------------------------------------------------------------

#define RNN_H 2350
#define RNN_E 768
#define RNN_O 19
#define RNN_B 64
#define RNN_T 256
#define RNN_EPS 1e-5f

typedef float v2f __attribute__((ext_vector_type(2)));
typedef float v8f __attribute__((ext_vector_type(8)));

__device__ __forceinline__ v8f rnn_wmma(v2f a, v2f b, v8f c) {
    return __builtin_amdgcn_wmma_f32_16x16x4_f32(
        false, a, false, b, (short)0, c, false, false);
}

// ---------------------------------------------------------------------------
// Generic WMMA GEMM: C(64 x N) = A1(64 x K1) * op(B1) + A2(64 x K2) * op(B2)
//                               + bias1 + bias2
// TRANSB=true : B stored row-major (N x K)  -> computes A @ B^T  (x @ W.T)
// TRANSB=false: B stored row-major (K x N)  -> computes A @ B
// One 16x16 C tile per wave; 8 waves/block arranged 4(M) x 2(N); M fixed 64.
// ---------------------------------------------------------------------------
template <bool TRANSB>
__global__ __launch_bounds__(256) void rnn_gemm64(
    const float* __restrict__ A1, int lda1, int K1,
    const float* __restrict__ A2, int lda2, int K2,
    const float* __restrict__ B1, int ldb1,
    const float* __restrict__ B2, int ldb2,
    const float* __restrict__ bias1, const float* __restrict__ bias2,
    float* __restrict__ C, int N)
{
    const int tid  = threadIdx.x;
    const int wid  = tid >> 5;
    const int lane = tid & 31;
    const int half = lane >> 4;     // 0: lanes 0-15, 1: lanes 16-31
    const int l16  = lane & 15;
    const int m0   = (wid & 3) << 4;                       // 0,16,32,48
    const int n0   = blockIdx.x * 32 + ((wid >> 2) << 4);  // 16-col tiles
    const int col  = n0 + l16;
    const int colc = (col < N) ? col : 0;  // clamp: garbage only hits unstored cols
    const int row  = m0 + l16;             // M=64 always fully valid
    const int h2   = half << 1;            // per-lane K sub-offset (0 or 2)

    v8f acc = {};

    auto run_seg = [&](const float* __restrict__ A, int lda, int K,
                       const float* __restrict__ Bw, int ldb) {
        // Per-lane base: element (row, h2) of A. Pairs (k, k+1) at even element
        // offsets from an 8B-aligned row base -> single b64 load each.
        const float* __restrict__ Arow = A + (size_t)row * lda + h2;
        const int K4 = K & ~3;

        if constexpr (TRANSB) {
            const float* __restrict__ Brow = Bw + (size_t)colc * ldb + h2;
#pragma unroll 4
            for (int k = 0; k < K4; k += 4) {
                const v2f a  = *(const v2f*)(Arow + k);
                const v2f bb = *(const v2f*)(Brow + k);
                acc = rnn_wmma(a, bb, acc);
            }
        } else {
#pragma unroll 4
            for (int k = 0; k < K4; k += 4) {
                const v2f a = *(const v2f*)(Arow + k);
                const int kb = k + h2;
                v2f bb;
                bb.x = Bw[(size_t)kb * ldb + colc];
                bb.y = Bw[(size_t)(kb + 1) * ldb + colc];
                acc = rnn_wmma(a, bb, acc);
            }
        }

        if (K4 < K) {  // tail (K % 4 != 0): guards confined to one iteration
            const int kb = K4 + h2;
            v2f a, bb;
            a.x = (kb     < K) ? Arow[K4]     : 0.0f;
            a.y = (kb + 1 < K) ? Arow[K4 + 1] : 0.0f;
            if constexpr (TRANSB) {
                const float* __restrict__ Brow = Bw + (size_t)colc * ldb + h2;
                bb.x = (kb     < K) ? Brow[K4]     : 0.0f;
                bb.y = (kb + 1 < K) ? Brow[K4 + 1] : 0.0f;
            } else {
                bb.x = (kb     < K) ? Bw[(size_t)kb * ldb + colc]       : 0.0f;
                bb.y = (kb + 1 < K) ? Bw[(size_t)(kb + 1) * ldb + colc] : 0.0f;
            }
            acc = rnn_wmma(a, bb, acc);
        }
    };

    run_seg(A1, lda1, K1, B1, ldb1);
    if (K2 > 0) run_seg(A2, lda2, K2, B2, ldb2);

    float bv = 0.0f;
    if (bias1) bv += bias1[colc];
    if (bias2) bv += bias2[colc];

    if (col < N) {
        for (int vi = 0; vi < 8; ++vi) {
            const int m = m0 + vi + (half << 3);  // C: VGPR vi = row vi / vi+8
            C[(size_t)m * N + col] = acc[vi] + bv;
        }
    }
}

// ---------------------------------------------------------------------------
// LayerNorm of input x: (B,T,E) row-major in -> transposed (T,B,E) out.
// ---------------------------------------------------------------------------
__global__ __launch_bounds__(256) void rnn_ln_x(
    const float* __restrict__ x, const float* __restrict__ g,
    const float* __restrict__ beta, float* __restrict__ xn_t)
{
    __shared__ float ssum[256];
    __shared__ float ssq[256];
    const int r   = blockIdx.x;        // r = b*T + t
    const int b   = r / RNN_T;
    const int t   = r % RNN_T;
    const int tid = threadIdx.x;
    const float* in  = x + (size_t)r * RNN_E;
    float*       out = xn_t + ((size_t)t * RNN_B + b) * RNN_E;

    float yv[3];
    float s = 0.0f, s2 = 0.0f;
    int cnt = 0;
    for (int i = tid; i < RNN_E; i += 256) {
        float v = in[i];
        yv[cnt++] = v;
        s += v; s2 += v * v;
    }
    ssum[tid] = s; ssq[tid] = s2;
    __syncthreads();
    for (int o = 128; o > 0; o >>= 1) {
        if (tid < o) { ssum[tid] += ssum[tid + o]; ssq[tid] += ssq[tid + o]; }
        __syncthreads();
    }
    const float mean = ssum[0] / RNN_E;
    const float var  = ssq[0] / RNN_E - mean * mean;
    const float rstd = rsqrtf(var + RNN_EPS);
    cnt = 0;
    for (int i = tid; i < RNN_E; i += 256) {
        out[i] = (yv[cnt++] - mean) * rstd * g[i] + beta[i];
    }
}

// ---------------------------------------------------------------------------
// tanh + LayerNorm + mask-blend.
//   hn = LN(tanh(pre)) ; state = m ? hn : state ; rawout=hn ; seqout=state
// seqout/rawout/seqlen may be null. rows = gridDim.x (= 64 here).
// ---------------------------------------------------------------------------
__global__ __launch_bounds__(256) void rnn_ln_blend(
    const float* __restrict__ pre, const float* __restrict__ g,
    const float* __restrict__ beta, float* __restrict__ state,
    float* __restrict__ rawout, float* __restrict__ seqout,
    int width, const int* __restrict__ seqlen, int t)
{
    __shared__ float ssum[256];
    __shared__ float ssq[256];
    const int r   = blockIdx.x;
    const int tid = threadIdx.x;
    const float* prow = pre + (size_t)r * width;

    float yv[10];  // ceil(2350/256) = 10
    float s = 0.0f, s2 = 0.0f;
    int cnt = 0;
    for (int i = tid; i < width; i += 256) {
        float v = tanhf(prow[i]);
        yv[cnt++] = v;
        s += v; s2 += v * v;
    }
    ssum[tid] = s; ssq[tid] = s2;
    __syncthreads();
    for (int o = 128; o > 0; o >>= 1) {
        if (tid < o) { ssum[tid] += ssum[tid + o]; ssq[tid] += ssq[tid + o]; }
        __syncthreads();
    }
    const float mean = ssum[0] / width;
    const float var  = ssq[0] / width - mean * mean;
    const float rstd = rsqrtf(var + RNN_EPS);

    bool keep = true;  // m == 1
    if (seqlen) keep = (t < seqlen[r]);

    cnt = 0;
    for (int i = tid; i < width; i += 256) {
        const float hn = (yv[cnt++] - mean) * rstd * g[i] + beta[i];
        const size_t idx = (size_t)r * width + i;
        if (rawout) rawout[idx] = hn;
        const float nv = keep ? hn : state[idx];
        state[idx] = nv;
        if (seqout) seqout[idx] = nv;
    }
}

// ---------------------------------------------------------------------------
// Attention for the single needed query row per batch:
//   sc[s] = scale * (u_b . hs[s,b,:]) ; a = softmax(sc) ;
//   wout[b,:] = sum_s a_s * hs[s,b,:]
// ---------------------------------------------------------------------------
__global__ __launch_bounds__(256) void rnn_attn(
    const float* __restrict__ u, const float* __restrict__ hs,
    float* __restrict__ wout, int D, float scale)
{
    __shared__ float sc[RNN_T];
    __shared__ float sred[256];
    const int b    = blockIdx.x;
    const int tid  = threadIdx.x;
    const int wid  = tid >> 5;
    const int lane = tid & 31;
    const float* ub = u + (size_t)b * D;

    for (int s = wid; s < RNN_T; s += 8) {
        const float* hrow = hs + ((size_t)s * RNN_B + b) * D;
        float d = 0.0f;
        for (int k = lane; k < D; k += 32) d += ub[k] * hrow[k];
        for (int o = 16; o > 0; o >>= 1) d += __shfl_xor(d, o, 32);
        if (lane == 0) sc[s] = d * scale;
    }
    __syncthreads();

    // softmax over 256 (blockDim == T)
    sred[tid] = sc[tid];
    __syncthreads();
    for (int o = 128; o > 0; o >>= 1) {
        if (tid < o) sred[tid] = fmaxf(sred[tid], sred[tid + o]);
        __syncthreads();
    }
    const float mx = sred[0];
    __syncthreads();
    const float e = expf(sc[tid] - mx);
    sred[tid] = e;
    __syncthreads();
    for (int o = 128; o > 0; o >>= 1) {
        if (tid < o) sred[tid] += sred[tid + o];
        __syncthreads();
    }
    const float inv = 1.0f / sred[0];
    __syncthreads();
    sc[tid] = e * inv;
    __syncthreads();

    for (int h = tid; h < D; h += 256) {
        float a = 0.0f;
        for (int s = 0; s < RNN_T; ++s)
            a += sc[s] * hs[((size_t)s * RNN_B + b) * D + h];
        wout[(size_t)b * D + h] = a;
    }
}

// ---------------------------------------------------------------------------
// Gather per-batch rows at t_b = seq_len[b]-1 from h1s, h2s, xn_t.
// ---------------------------------------------------------------------------
__global__ __launch_bounds__(256) void rnn_gather(
    const float* __restrict__ h1s, const float* __restrict__ h2s,
    const float* __restrict__ xn_t, const int* __restrict__ seqlen,
    float* __restrict__ qin1, float* __restrict__ qin2,
    float* __restrict__ xg)
{
    const int b = blockIdx.x;
    int tb = seqlen[b] - 1;
    tb = (tb < 0) ? 0 : ((tb > RNN_T - 1) ? RNN_T - 1 : tb);
    const int tid = threadIdx.x;
    const float* s1 = h1s  + ((size_t)tb * RNN_B + b) * RNN_H;
    const float* s2 = h2s  + ((size_t)tb * RNN_B + b) * RNN_E;
    const float* s3 = xn_t + ((size_t)tb * RNN_B + b) * RNN_E;
    for (int i = tid; i < RNN_H; i += 256) qin1[(size_t)b * RNN_H + i] = s1[i];
    for (int i = tid; i < RNN_E; i += 256) {
        qin2[(size_t)b * RNN_E + i] = s2[i];
        xg  [(size_t)b * RNN_E + i] = s3[i];
    }
}

// ---------------------------------------------------------------------------
// logits = clip(h2f @ Wc^T + bc, -10, 10)   (64 x 19, K=768)
// ---------------------------------------------------------------------------
__global__ __launch_bounds__(256) void rnn_classifier(
    const float* __restrict__ h2f, const float* __restrict__ Wc,
    const float* __restrict__ bc, float* __restrict__ out)
{
    const int idx = blockIdx.x * 256 + threadIdx.x;
    if (idx >= RNN_B * RNN_O) return;
    const int b = idx / RNN_O;
    const int o = idx % RNN_O;
    const float* hr = h2f + (size_t)b * RNN_E;
    const float* wr = Wc + (size_t)o * RNN_E;
    float d = bc[o];
    for (int k = 0; k < RNN_E; ++k) d += hr[k] * wr[k];
    d = fminf(10.0f, fmaxf(-10.0f, d));
    out[idx] = d;
}

// ---------------------------------------------------------------------------
extern "C" void kernel_launch(void* const* d_in, const int* in_sizes, int n_in,
                              void* d_out, int out_size, void* d_ws, size_t ws_size,
                              hipStream_t stream)
{
    (void)in_sizes; (void)n_in; (void)out_size; (void)ws_size;
    const int H = RNN_H, E = RNN_E, B = RNN_B, T = RNN_T;

    const float* x     = (const float*)d_in[0];
    const int*   seq   = (const int*)  d_in[1];
    const float* in_g  = (const float*)d_in[2];
    const float* in_b  = (const float*)d_in[3];
    const float* Wih1  = (const float*)d_in[4];
    const float* bih1  = (const float*)d_in[5];
    const float* Whh1  = (const float*)d_in[6];
    const float* bhh1  = (const float*)d_in[7];
    const float* ln1g  = (const float*)d_in[8];
    const float* ln1b  = (const float*)d_in[9];
    const float* Wih2  = (const float*)d_in[10];
    const float* bih2  = (const float*)d_in[11];
    const float* Whh2  = (const float*)d_in[12];
    const float* bhh2  = (const float*)d_in[13];
    const float* ln2g  = (const float*)d_in[14];
    const float* ln2b  = (const float*)d_in[15];
    const float* Wqkv1 = (const float*)d_in[16];
    const float* bqkv1 = (const float*)d_in[17];
    const float* Wo1   = (const float*)d_in[18];
    const float* bo1   = (const float*)d_in[19];
    const float* Wqkv2 = (const float*)d_in[20];
    const float* bqkv2 = (const float*)d_in[21];
    const float* Wo2   = (const float*)d_in[22];
    const float* bo2   = (const float*)d_in[23];
    const float* Wc    = (const float*)d_in[24];
    const float* bc    = (const float*)d_in[25];

    float* ws = (float*)d_ws;
    size_t off = 0;
    auto alloc = [&](size_t n) { float* p = ws + off; off += n; return p; };

    float* xn_t = alloc((size_t)T * B * E);   // LN(x), transposed (T,B,E)
    float* h1s  = alloc((size_t)T * B * H);   // scan1 layer-1 states
    float* h2s  = alloc((size_t)T * B * E);   // scan1 layer-2 states
    float* h1   = alloc((size_t)B * H);
    float* h2   = alloc((size_t)B * E);
    float* h1n  = alloc((size_t)B * H);       // pre-blend LN output (layer 1)
    float* pre1 = alloc((size_t)B * H);
    float* pre2 = alloc((size_t)B * E);
    float* qin1 = alloc((size_t)B * H);
    float* qin2 = alloc((size_t)B * E);
    float* xg   = alloc((size_t)B * E);
    float* q1   = alloc((size_t)B * H);
    float* u1   = alloc((size_t)B * H);
    float* w1   = alloc((size_t)B * H);
    float* ctx1 = alloc((size_t)B * H);
    float* out1 = alloc((size_t)B * H);
    float* q2   = alloc((size_t)B * E);
    float* u2   = alloc((size_t)B * E);
    float* w2   = alloc((size_t)B * E);
    float* ctx2 = alloc((size_t)B * E);
    float* out2 = alloc((size_t)B * E);
    float* preF1 = alloc((size_t)B * H);
    float* h1nF  = alloc((size_t)B * H);
    float* preF2 = alloc((size_t)B * E);
    float* h2nF  = alloc((size_t)B * E);

    const int gH = (H + 31) / 32;  // 74
    const int gE = (E + 31) / 32;  // 24

    (void)hipMemsetAsync(h1, 0, (size_t)B * H * sizeof(float), stream);
    (void)hipMemsetAsync(h2, 0, (size_t)B * E * sizeof(float), stream);

    // Input LayerNorm, transposed to (T,B,E)
    rnn_ln_x<<<B * T, 256, 0, stream>>>(x, in_g, in_b, xn_t);

    // -------- scan 1 (truly sequential part) --------
    for (int t = 0; t < T; ++t) {
        const float* xt = xn_t + (size_t)t * B * E;
        rnn_gemm64<true><<<gH, 256, 0, stream>>>(
            xt, E, E, h1, H, H, Wih1, E, Whh1, H, bih1, bhh1, pre1, H);
        rnn_ln_blend<<<B, 256, 0, stream>>>(
            pre1, ln1g, ln1b, h1, h1n, h1s + (size_t)t * B * H, H, seq, t);
        rnn_gemm64<true><<<gE, 256, 0, stream>>>(
            h1n, H, H, h2, E, E, Wih2, H, Whh2, E, bih2, bhh2, pre2, E);
        rnn_ln_blend<<<B, 256, 0, stream>>>(
            pre2, ln2g, ln2b, h2, nullptr, h2s + (size_t)t * B * E, E, seq, t);
    }

    // -------- gather the single needed row per batch --------
    rnn_gather<<<B, 256, 0, stream>>>(h1s, h2s, xn_t, seq, qin1, qin2, xg);

    // -------- MHA1 (d = H), K/V folded --------
    rnn_gemm64<true><<<gH, 256, 0, stream>>>(
        qin1, H, H, nullptr, 0, 0, Wqkv1, H, nullptr, 0, bqkv1, nullptr, q1, H);
    rnn_gemm64<false><<<gH, 256, 0, stream>>>(   // u = q @ Wk  (Wk^T q)
        q1, H, H, nullptr, 0, 0, Wqkv1 + (size_t)H * H, H, nullptr, 0,
        nullptr, nullptr, u1, H);
    rnn_attn<<<B, 256, 0, stream>>>(u1, h1s, w1, H, 1.0f / sqrtf((float)H));
    rnn_gemm64<true><<<gH, 256, 0, stream>>>(    // ctx = Wv w + bv
        w1, H, H, nullptr, 0, 0, Wqkv1 + (size_t)2 * H * H, H, nullptr, 0,
        bqkv1 + 2 * H, nullptr, ctx1, H);
    rnn_gemm64<true><<<gH, 256, 0, stream>>>(    // out = ctx @ Wo^T + bo
        ctx1, H, H, nullptr, 0, 0, Wo1, H, nullptr, 0, bo1, nullptr, out1, H);

    // -------- MHA2 (d = E), K/V folded --------
    rnn_gemm64<true><<<gE, 256, 0, stream>>>(
        qin2, E, E, nullptr, 0, 0, Wqkv2, E, nullptr, 0, bqkv2, nullptr, q2, E);
    rnn_gemm64<false><<<gE, 256, 0, stream>>>(
        q2, E, E, nullptr, 0, 0, Wqkv2 + (size_t)E * E, E, nullptr, 0,
        nullptr, nullptr, u2, E);
    rnn_attn<<<B, 256, 0, stream>>>(u2, h2s, w2, E, 1.0f / sqrtf((float)E));
    rnn_gemm64<true><<<gE, 256, 0, stream>>>(
        w2, E, E, nullptr, 0, 0, Wqkv2 + (size_t)2 * E * E, E, nullptr, 0,
        bqkv2 + 2 * E, nullptr, ctx2, E);
    rnn_gemm64<true><<<gE, 256, 0, stream>>>(
        ctx2, E, E, nullptr, 0, 0, Wo2, E, nullptr, 0, bo2, nullptr, out2, E);

    // -------- collapsed scan 2: one cell eval at t_b per batch --------
    rnn_gemm64<true><<<gH, 256, 0, stream>>>(
        xg, E, E, out1, H, H, Wih1, E, Whh1, H, bih1, bhh1, preF1, H);
    rnn_ln_blend<<<B, 256, 0, stream>>>(
        preF1, ln1g, ln1b, h1nF, nullptr, nullptr, H, nullptr, 0);
    rnn_gemm64<true><<<gE, 256, 0, stream>>>(
        h1nF, H, H, out2, E, E, Wih2, H, Whh2, E, bih2, bhh2, preF2, E);
    rnn_ln_blend<<<B, 256, 0, stream>>>(
        preF2, ln2g, ln2b, h2nF, nullptr, nullptr, E, nullptr, 0);

    // -------- classifier --------
    rnn_classifier<<<(B * RNN_O + 255) / 256, 256, 0, stream>>>(
        h2nF, Wc, bc, (float*)d_out);
}